// DMaSIFSegment_15942918603396
// MI455X (gfx1250) — compile-verified
//
#include <hip/hip_runtime.h>
#include <hip/hip_bf16.h>

typedef float v2f __attribute__((ext_vector_type(2)));
typedef float v8f __attribute__((ext_vector_type(8)));
typedef int   v4i __attribute__((vector_size(16)));
typedef __attribute__((address_space(1))) v4i v4i_g;   // global (AS1)
typedef __attribute__((address_space(3))) v4i v4i_l;   // LDS (AS3)

#define NPTS   2048
#define OCH    64
#define PSCALE 0.0785674201318385f   // 1/(sqrt(2)*9)

#if __has_builtin(__builtin_amdgcn_global_load_async_to_lds_b128)
#define HAVE_ASYNC_LDS 1
#else
#define HAVE_ASYNC_LDS 0
#endif

__device__ __forceinline__ void wait_asynccnt0() {
#if __has_builtin(__builtin_amdgcn_s_wait_asynccnt)
  __builtin_amdgcn_s_wait_asynccnt(0);
#else
  asm volatile("s_wait_asynccnt 0" ::: "memory");
#endif
}

// ---------------------------------------------------------------------------
// Generic fused linear: out = act(A @ W^T + bias) [+ add]
// A: 2048 x K (row major), W: 64 x K, out: 2048 x 64.
// One wave computes one 16x16 tile via chained V_WMMA_F32_16X16X4_F32.
// mode: 0 = none, 1 = relu, 2 = leaky_relu(0.2)
// ---------------------------------------------------------------------------
__global__ __launch_bounds__(256) void k_gemm(
    const float* __restrict__ A, int K,
    const float* __restrict__ W, const float* __restrict__ bias,
    const float* __restrict__ add, float* __restrict__ out, int mode) {
  const int lane = threadIdx.x & 31;
  const int wv   = threadIdx.x >> 5;
  const int tile = blockIdx.x * 8 + wv;      // 512 tiles: 128 row x 4 col
  const int m0   = (tile >> 2) << 4;
  const int n0   = (tile & 3) << 4;
  const int half = lane >> 4;                // 0: lanes 0-15, 1: lanes 16-31
  const int l16  = lane & 15;
  const int koff = half << 1;                // A/B K offset per lane-half

  const float* Arow = A + (size_t)(m0 + l16) * K;
  const float* Wrow = W + (size_t)(n0 + l16) * K;

  v8f acc = {0.f, 0.f, 0.f, 0.f, 0.f, 0.f, 0.f, 0.f};
  for (int k0 = 0; k0 < K; k0 += 4) {
    v2f a = *(const v2f*)(Arow + k0 + koff);   // VGPR0=K(koff), VGPR1=K(koff+1)
    v2f b = *(const v2f*)(Wrow + k0 + koff);   // B[k][n] = W[n][k]
    acc = __builtin_amdgcn_wmma_f32_16x16x4_f32(
        false, a, false, b, (short)0, acc, false, false);
  }

  const int   c  = n0 + l16;
  const float bv = bias[c];
#pragma unroll
  for (int r = 0; r < 8; ++r) {
    const int m = m0 + r + half * 8;
    float v = acc[r] + bv;
    if (mode == 1)      v = fmaxf(v, 0.f);
    else if (mode == 2) v = (v > 0.f) ? v : 0.2f * v;
    if (add) v += add[m * OCH + c];
    out[m * OCH + c] = v;
  }
}

// ---------------------------------------------------------------------------
// Group norm matching the reference: group g = channels [16g,16g+16),
// statistics over all 2048 rows (biased variance).
// ---------------------------------------------------------------------------
__global__ __launch_bounds__(256) void k_gn(
    const float* __restrict__ x, const float* __restrict__ w,
    const float* __restrict__ b, float* __restrict__ out) {
  __shared__ float red[256], red2[256];
  const int g   = blockIdx.x;
  const int tid = threadIdx.x;
  const int c0  = g * 16;
  float s = 0.f, s2 = 0.f;
  for (int e = tid; e < NPTS * 16; e += 256) {
    const int n = e >> 4, c = c0 + (e & 15);
    const float v = x[n * OCH + c];
    s += v; s2 += v * v;
  }
  red[tid] = s; red2[tid] = s2;
  __syncthreads();
  for (int off = 128; off > 0; off >>= 1) {
    if (tid < off) { red[tid] += red[tid + off]; red2[tid] += red2[tid + off]; }
    __syncthreads();
  }
  const float mean = red[0] * (1.f / 32768.f);
  const float var  = red2[0] * (1.f / 32768.f) - mean * mean;
  const float rs   = rsqrtf(var + 1e-5f);
  for (int e = tid; e < NPTS * 16; e += 256) {
    const int n = e >> 4, c = c0 + (e & 15);
    out[n * OCH + c] = (x[n * OCH + c] - mean) * rs * w[c] + b[c];
  }
}

// ---------------------------------------------------------------------------
// interact(): per output point i, over all n:
//   w   = exp(-|d|^2 (2-dot)^2)
//   h1  = relu(w1 @ (nuv_i . diff) + b1)              (8, VALU)
//   h2  = relu(h1 @ w2^T + b2)                        (64, two fp32 WMMAs/ctile)
//   out[i] += w * feats[n] * h2                       (elementwise, VALU)
// Block = 8 waves x 4 points; feats/points/normals staged per 128-n chunk in
// LDS (row stride 68 -> conflict-free half-wave access). Feats staged via
// async global->LDS when available; next chunk prefetched during compute.
// ---------------------------------------------------------------------------
#define CHUNK 128
#define FSTR  68

__global__ __launch_bounds__(256) void k_interact(
    const float* __restrict__ points, const float* __restrict__ nuv,
    const float* __restrict__ feats,
    const float* __restrict__ cw1, const float* __restrict__ cb1,
    const float* __restrict__ cw2, const float* __restrict__ cb2,
    float* __restrict__ out) {
  __shared__ __attribute__((aligned(16))) float s_feats[CHUNK * FSTR];
  __shared__ float s_pts[CHUNK * 3];
  __shared__ float s_nrm[CHUNK * 3];

  const int tid  = threadIdx.x;
  const int lane = tid & 31;
  const int wv   = tid >> 5;
  const int half = lane >> 4;
  const int l16  = lane & 15;
  const int koff = half << 1;
  const int i_base = blockIdx.x * 32 + wv * 4;

  // Loop-invariant B tiles: B[k][c] = w2[c][k]  (w2 is 64x8)
  v2f b0[4], b1[4];
  float b2v[4];
#pragma unroll
  for (int cb = 0; cb < 4; ++cb) {
    const int c = cb * 16 + l16;
    b0[cb][0] = cw2[c * 8 + koff];
    b0[cb][1] = cw2[c * 8 + koff + 1];
    b1[cb][0] = cw2[c * 8 + 4 + koff];
    b1[cb][1] = cw2[c * 8 + 4 + koff + 1];
    b2v[cb]   = cb2[c];
  }
  float w1r[24], b1r[8];
#pragma unroll
  for (int k = 0; k < 24; ++k) w1r[k] = cw1[k];
#pragma unroll
  for (int k = 0; k < 8; ++k)  b1r[k] = cb1[k];

  float acc[4][4];
#pragma unroll
  for (int ii = 0; ii < 4; ++ii)
#pragma unroll
    for (int cb = 0; cb < 4; ++cb) acc[ii][cb] = 0.f;

  for (int chunk0 = 0; chunk0 < NPTS; chunk0 += CHUNK) {
    // ---- stage feats chunk (128 x 64 floats) + points/normals into LDS ----
#if HAVE_ASYNC_LDS
#pragma unroll
    for (int k = 0; k < 8; ++k) {
      const int idx = tid + k * 256;            // 0..2047 float4 slots
      const int row = idx >> 4, c4 = (idx & 15) << 2;
      float* fsrc = const_cast<float*>(feats) + (size_t)(chunk0 + row) * OCH + c4;
      __builtin_amdgcn_global_load_async_to_lds_b128(
          (v4i_g*)fsrc, (v4i_l*)&s_feats[row * FSTR + c4], 0, 0);
    }
#else
#pragma unroll
    for (int k = 0; k < 8; ++k) {
      const int idx = tid + k * 256;            // 0..2047 float4 slots
      const int row = idx >> 4, c4 = (idx & 15) << 2;
      const float4 v = *(const float4*)(feats + (size_t)(chunk0 + row) * OCH + c4);
      *(float4*)(s_feats + row * FSTR + c4) = v;
    }
#endif
    if (tid < CHUNK) {
      const int n = chunk0 + tid;
      s_pts[tid * 3 + 0] = points[n * 3 + 0] * PSCALE;
      s_pts[tid * 3 + 1] = points[n * 3 + 1] * PSCALE;
      s_pts[tid * 3 + 2] = points[n * 3 + 2] * PSCALE;
      s_nrm[tid * 3 + 0] = nuv[n * 9 + 0];
      s_nrm[tid * 3 + 1] = nuv[n * 9 + 1];
      s_nrm[tid * 3 + 2] = nuv[n * 9 + 2];
    }
#if HAVE_ASYNC_LDS
    wait_asynccnt0();
#endif
    __syncthreads();

    // prefetch next chunk's feats into cache while we compute this one
    if (chunk0 + CHUNK < NPTS) {
      const float* nf = feats + (size_t)(chunk0 + CHUNK) * OCH;
      __builtin_prefetch(nf + tid * 32, 0, 0);   // 256 threads x 128B lines = 32KB
    }

#pragma unroll 1
    for (int ii = 0; ii < 4; ++ii) {
      const int i = i_base + ii;
      const float xi0 = points[i * 3 + 0] * PSCALE;
      const float xi1 = points[i * 3 + 1] * PSCALE;
      const float xi2 = points[i * 3 + 2] * PSCALE;
      const float nv0 = nuv[i * 9 + 0], nv1 = nuv[i * 9 + 1], nv2 = nuv[i * 9 + 2];
      const float nv3 = nuv[i * 9 + 3], nv4 = nuv[i * 9 + 4], nv5 = nuv[i * 9 + 5];
      const float nv6 = nuv[i * 9 + 6], nv7 = nuv[i * 9 + 7], nv8 = nuv[i * 9 + 8];

#pragma unroll 1
      for (int t = 0; t < CHUNK / 16; ++t) {
        // per-lane pair math for n = chunk0 + t*16 + (lane&15)
        const int nl = t * 16 + l16;
        const float dx = s_pts[nl * 3 + 0] - xi0;
        const float dy = s_pts[nl * 3 + 1] - xi1;
        const float dz = s_pts[nl * 3 + 2] - xi2;
        const float dot = nv0 * s_nrm[nl * 3 + 0] + nv1 * s_nrm[nl * 3 + 1] +
                          nv2 * s_nrm[nl * 3 + 2];
        const float q  = 2.f - dot;
        const float d2 = (dx * dx + dy * dy + dz * dz) * q * q;
        const float wgt = __expf(-d2);
        const float X0 = nv0 * dx + nv1 * dy + nv2 * dz;
        const float X1 = nv3 * dx + nv4 * dy + nv5 * dz;
        const float X2 = nv6 * dx + nv7 * dy + nv8 * dz;
        float h1[8];
#pragma unroll
        for (int j = 0; j < 8; ++j)
          h1[j] = fmaxf(w1r[j * 3] * X0 + w1r[j * 3 + 1] * X1 +
                        w1r[j * 3 + 2] * X2 + b1r[j], 0.f);

        // A tiles (16 n-rows x K=8, split into two K=4 tiles)
        v2f a0, a1;
        a0[0] = h1[koff];     a0[1] = h1[koff + 1];
        a1[0] = h1[4 + koff]; a1[1] = h1[4 + koff + 1];

        // redistribute w: lane needs w for n-rows r + 8*half
        float wv8[8];
#pragma unroll
        for (int r = 0; r < 8; ++r)
          wv8[r] = __shfl(wgt, r + half * 8, 32);

#pragma unroll
        for (int cb = 0; cb < 4; ++cb) {
          v8f cmat = {0.f, 0.f, 0.f, 0.f, 0.f, 0.f, 0.f, 0.f};
          cmat = __builtin_amdgcn_wmma_f32_16x16x4_f32(
              false, a0, false, b0[cb], (short)0, cmat, false, false);
          cmat = __builtin_amdgcn_wmma_f32_16x16x4_f32(
              false, a1, false, b1[cb], (short)0, cmat, false, false);
          const int cc = cb * 16 + l16;
          float a = 0.f;
#pragma unroll
          for (int r = 0; r < 8; ++r) {
            const int nlr = t * 16 + r + half * 8;
            const float h2 = fmaxf(cmat[r] + b2v[cb], 0.f);
            a += wv8[r] * s_feats[nlr * FSTR + cc] * h2;
          }
          acc[ii][cb] += a;
        }
      }
    }
    __syncthreads();
  }

#pragma unroll
  for (int ii = 0; ii < 4; ++ii) {
#pragma unroll
    for (int cb = 0; cb < 4; ++cb) {
      const float v = acc[ii][cb] + __shfl_xor(acc[ii][cb], 16, 32);
      if (half == 0) out[(i_base + ii) * OCH + cb * 16 + l16] = v;
    }
  }
}

// ---------------------------------------------------------------------------
// Host orchestration: 4 layers, ~10 kernels each, all on `stream`.
// Param order per layer (22 arrays): in_w1,in_b1,in_w2,in_b2,gn_in_w,gn_in_b,
// conv_w1,conv_b1,conv_w2,conv_b2,out_w1,out_b1,out_w2,out_b2,gn_out_w,
// gn_out_b,olay_w1,olay_b1,olay_w2,olay_b2,ilay_w,ilay_b
// ---------------------------------------------------------------------------
extern "C" void kernel_launch(void* const* d_in, const int* in_sizes, int n_in,
                              void* d_out, int out_size, void* d_ws, size_t ws_size,
                              hipStream_t stream) {
  (void)in_sizes; (void)n_in; (void)out_size; (void)ws_size;
  const float* features = (const float*)d_in[0];
  const float* points   = (const float*)d_in[1];
  const float* nuv      = (const float*)d_in[2];
  // d_in[3] = ranges (unused)

  float* ws = (float*)d_ws;
  const size_t NF = (size_t)NPTS * OCH;
  float* bufA = ws;
  float* bufB = ws + NF;
  float* xa   = ws + 2 * NF;
  float* xb   = ws + 3 * NF;

  const dim3 gg(64), bb(256), gn(4);
  const float* xcur = features;
  int K_in = 16;

  for (int l = 0; l < 4; ++l) {
    const float* const* P = (const float* const*)(d_in + 4 + l * 22);

    // f = lrelu(x @ in_w1^T + b); f = lrelu(f @ in_w2^T + b); group_norm
    hipLaunchKernelGGL(k_gemm, gg, bb, 0, stream, xcur, K_in, P[0], P[1],
                       (const float*)nullptr, bufA, 2);
    hipLaunchKernelGGL(k_gemm, gg, bb, 0, stream, bufA, OCH, P[2], P[3],
                       (const float*)nullptr, bufB, 2);
    hipLaunchKernelGGL(k_gn, gn, bb, 0, stream, bufB, P[4], P[5], bufA);

    // quasi-geodesic interaction (N^2 WMMA core)
    hipLaunchKernelGGL(k_interact, gg, bb, 0, stream, points, nuv, bufA,
                       P[6], P[7], P[8], P[9], bufB);

    // f = lrelu(f @ out_w1^T); f = lrelu(f @ out_w2^T); group_norm
    hipLaunchKernelGGL(k_gemm, gg, bb, 0, stream, bufB, OCH, P[10], P[11],
                       (const float*)nullptr, bufA, 2);
    hipLaunchKernelGGL(k_gemm, gg, bb, 0, stream, bufA, OCH, P[12], P[13],
                       (const float*)nullptr, bufB, 2);
    hipLaunchKernelGGL(k_gn, gn, bb, 0, stream, bufB, P[14], P[15], bufA);

    // h = relu(f @ olay_w1^T); h = h @ olay_w2^T; x = x @ ilay_w^T + h
    hipLaunchKernelGGL(k_gemm, gg, bb, 0, stream, bufA, OCH, P[16], P[17],
                       (const float*)nullptr, bufB, 1);
    hipLaunchKernelGGL(k_gemm, gg, bb, 0, stream, bufB, OCH, P[18], P[19],
                       (const float*)nullptr, bufA, 0);

    float* xnext = (l == 3) ? (float*)d_out : ((l & 1) ? xb : xa);
    hipLaunchKernelGGL(k_gemm, gg, bb, 0, stream, xcur, K_in, P[20], P[21],
                       bufA, xnext, 0);
    xcur = xnext;
    K_in = OCH;
  }
}